// ModelNew_23656679867276
// MI455X (gfx1250) — compile-verified
//
#include <hip/hip_runtime.h>
#include <cstdint>

typedef __attribute__((ext_vector_type(2))) float v2f;
typedef __attribute__((ext_vector_type(8))) float v8f;

#define WAVES_PER_BLOCK 8
#define TILE 256            // floats per tile (16x16)
#define TILE_BYTES 1024

// One wave handles one full row (32768 f32) sequentially in 16x16 tiles.
// Tile pipeline: async global->LDS copy (double buffered, ASYNCcnt) ->
// WMMA matmul-scan (P = T x UpperTriOnes via 4 chained 16x16x4 f32 WMMAs) ->
// shuffle-based cross-row offsets -> add serial carry -> store.
__global__ __launch_bounds__(256) void cumsum_rows_kernel(
    const float* __restrict__ x, float* __restrict__ out, int rows, int cols) {
  __shared__ float lds[WAVES_PER_BLOCK][2][TILE];   // 8 waves * 2 buffers * 1KB = 16KB

  const int lane = threadIdx.x & 31;
  const int wave = threadIdx.x >> 5;
  const int row  = blockIdx.x * WAVES_PER_BLOCK + wave;
  if (row >= rows) return;   // wave-uniform: EXEC stays all-ones per active wave

  const float* xrow = x + (size_t)row * (size_t)cols;
  float*       orow = out + (size_t)row * (size_t)cols;

  const int n    = lane & 15;   // N (column) index in C/D layout
  const int half = lane >> 4;   // lane half selects M/K groups

  // LDS byte address of this wave's buffers (+ per-lane 16B slice for async dst).
  // Flat shared-aperture addresses carry the LDS offset in the low 32 bits.
  const unsigned ldsAddr[2] = {
    (unsigned)(uintptr_t)(&lds[wave][0][0]) + 16u * (unsigned)lane,
    (unsigned)(uintptr_t)(&lds[wave][1][0]) + 16u * (unsigned)lane
  };

  // B operand: U[4k..4k+3, :], U[i][j] = (i <= j). f32 B layout mirrors A:
  // v0 = rows {4k, 4k+2} across lane halves, v1 = rows {4k+1, 4k+3}.
  v2f Bk[4];
#pragma unroll
  for (int k = 0; k < 4; ++k) {
    const int kr = 4 * k + 2 * half;
    Bk[k].x = (kr     <= n) ? 1.0f : 0.0f;
    Bk[k].y = (kr + 1 <= n) ? 1.0f : 0.0f;
  }

  const int ntiles = cols / TILE;

  // Async copy of one 1KB tile: 2 x b128 per lane, contiguous 512B per instr.
  // INST_OFFSET applies to both the LDS dst and the global src address.
  auto issue_tile = [&](int t, int buf) {
    const unsigned gvo = (unsigned)t * TILE_BYTES + 16u * (unsigned)lane;
    asm volatile(
        "global_load_async_to_lds_b128 %0, %1, %2 offset:0\n\t"
        "global_load_async_to_lds_b128 %0, %1, %2 offset:512"
        :
        : "v"(ldsAddr[buf]), "v"(gvo), "s"(xrow)
        : "memory");
  };

  issue_tile(0, 0);
  float carry = 0.0f;

  for (int t = 0; t < ntiles; ++t) {
    const int buf = t & 1;
    if (t + 1 < ntiles) {
      issue_tile(t + 1, buf ^ 1);
      asm volatile("s_wait_asynccnt 0x2" ::: "memory");  // tile t's 2 copies done
    } else {
      asm volatile("s_wait_asynccnt 0x0" ::: "memory");
    }

    // A operand: T[:, 4k:4k+4]; lane reads its float2 (K pair) from LDS.
    const v2f* l2 = (const v2f*)&lds[wave][buf][0];
    const int m = n;  // A-matrix row index for this lane
    v8f acc = {};     // C starts at 0; carry/offsets added after
#pragma unroll
    for (int k = 0; k < 4; ++k) {
      v2f a = l2[(unsigned)(16 * m + 4 * k + 2 * half) >> 1];
      acc = __builtin_amdgcn_wmma_f32_16x16x4_f32(
          false, a, false, Bk[k], (short)0, acc, false, false);
    }

    // acc = P (within-row inclusive prefix). Column 15 (row totals) lives in
    // lanes 15 (M=0..7 across VGPRs) and 31 (M=8..15). Exclusive-scan the 8
    // per-lane values, then broadcast.
    float e[9];
    e[0] = 0.0f;
#pragma unroll
    for (int v = 0; v < 8; ++v) e[v + 1] = e[v] + acc[v];

    const float tot_lo = __shfl(e[8], 15, 32);           // sum of rows 0..7
    const int   src    = half ? 31 : 15;
    const float base_add = carry + (half ? tot_lo : 0.0f);

    const int tb = t * TILE;
#pragma unroll
    for (int v = 0; v < 8; ++v) {
      const float offv = __shfl(e[v], src, 32) + base_add;
      // element (M = v + 8*half, N = n) -> tb + 16*M + N
      orow[tb + 16 * v + 128 * half + n] = acc[v] + offv;
    }
    carry += tot_lo + __shfl(e[8], 31, 32);
  }
}

extern "C" void kernel_launch(void* const* d_in, const int* in_sizes, int n_in,
                              void* d_out, int out_size, void* d_ws, size_t ws_size,
                              hipStream_t stream) {
  (void)n_in; (void)d_ws; (void)ws_size; (void)out_size;
  const float* x = (const float*)d_in[0];
  float* out = (float*)d_out;

  const int cols = 32768;
  const int rows = in_sizes[0] / cols;

  dim3 block(32 * WAVES_PER_BLOCK);
  dim3 grid((rows + WAVES_PER_BLOCK - 1) / WAVES_PER_BLOCK);
  hipLaunchKernelGGL(cumsum_rows_kernel, grid, block, 0, stream, x, out, rows, cols);
}